// NonMaximaSuppression3d_37434934952182
// MI455X (gfx1250) — compile-verified
//
#include <hip/hip_runtime.h>

// ---------------------------------------------------------------------------
// 3D NMS (3x3x3, center-excluded, replicate padding, max clamped at 0).
// Memory-bound: 420 MB HBM traffic floor (~18 us @ 23.3 TB/s). Strategy:
// LDS-tiled stencil fed by CDNA5 async global->LDS DMA with a depth-2
// software pipeline (ASYNCcnt), nontemporal stores for the output.
// ---------------------------------------------------------------------------

namespace {
constexpr int Dd = 32, Hh = 160, Ww = 160;
constexpr int TH = 16;                  // rows per block tile
constexpr int NTH = Hh / TH;            // 10
constexpr int DCHUNK = 16;              // depths per block
constexpr int NTD = Dd / DCHUNK;        // 2
constexpr int HWP = Ww + 2;             // 162 (padded row length)
constexpr int HHP = TH + 2;             // 18  (padded row count)
constexpr int PLANE = HHP * HWP;        // 2916 floats per halo plane
constexpr int NTHREADS = 256;           // 8 wave32s
constexpr int LOADS = (PLANE + NTHREADS - 1) / NTHREADS;  // 12 (uniform per lane)
constexpr int OUTS  = (TH * Ww) / NTHREADS;               // 10 outputs per thread
}

#if defined(__has_builtin)
#if __has_builtin(__builtin_amdgcn_global_load_async_to_lds_b32)
#define NMS_ASYNC 1
#endif
#endif

typedef __attribute__((address_space(1))) int* glb_i32p;
typedef __attribute__((address_space(3))) int* lds_i32p;

__global__ __launch_bounds__(NTHREADS) void nms3d_kernel(const float* __restrict__ x,
                                                         float* __restrict__ out) {
  __shared__ float sm[4][PLANE];        // 46,656 B of the 320 KB WGP LDS pool

  int bid = blockIdx.x;
  const int tileH = bid % NTH; bid /= NTH;
  const int tileD = bid % NTD; bid /= NTD;
  const int slice = bid;                // b*CH + ch

  const size_t sliceOff = (size_t)slice * (Dd * Hh * Ww);
  const float* __restrict__ xs = x + sliceOff;
  float* __restrict__ os = out + sliceOff;

  const int h0 = tileH * TH;
  const int d0 = tileD * DCHUNK;
  const int t  = (int)threadIdx.x;

  // Precompute this thread's LDS slots and replicate-clamped global offsets.
  // Tail lanes duplicate the last element so every lane issues exactly LOADS
  // async ops per plane (needed for the s_wait_asynccnt<=LOADS pipeline).
  int ldsIdx[LOADS];
  int gOff[LOADS];
#pragma unroll
  for (int j = 0; j < LOADS; ++j) {
    int i = t + j * NTHREADS;
    if (i >= PLANE) i = PLANE - 1;
    ldsIdx[j] = i;
    const int hh = i / HWP - 1;
    const int ww = i % HWP - 1;
    const int gh = min(max(h0 + hh, 0), Hh - 1);
    const int gw = min(max(ww, 0), Ww - 1);
    gOff[j] = gh * Ww + gw;
  }

  auto issuePlane = [&](int d, int buf) {
    d = min(max(d, 0), Dd - 1);
    const float* __restrict__ p = xs + (size_t)d * (Hh * Ww);
#pragma unroll
    for (int j = 0; j < LOADS; ++j) {
#if NMS_ASYNC
      __builtin_amdgcn_global_load_async_to_lds_b32(
          (glb_i32p)(p + gOff[j]),
          (lds_i32p)&sm[buf][ldsIdx[j]],
          0, 0);
#else
      sm[buf][ldsIdx[j]] = p[gOff[j]];
#endif
    }
  };

  // Prologue: planes d0-1, d0, d0+1 in flight.
  issuePlane(d0 - 1, 0);
  issuePlane(d0,     1);
  issuePlane(d0 + 1, 2);

  for (int k = 0; k < DCHUNK; ++k) {
    const int d = d0 + k;

    // Keep plane d+2 in flight while we compute depth d.
    issuePlane(d + 2, (k + 3) & 3);
#if NMS_ASYNC
    // Async loads complete in order and each lane issued exactly LOADS ops
    // per plane, so <=LOADS outstanding means only the just-issued plane may
    // still be in flight; planes d-1, d, d+1 are resident.
    asm volatile("s_wait_asynccnt %0" :: "n"(LOADS) : "memory");
#endif
    __syncthreads();

    const float* __restrict__ pp = sm[(k + 0) & 3];   // depth d-1
    const float* __restrict__ pc = sm[(k + 1) & 3];   // depth d
    const float* __restrict__ pn = sm[(k + 2) & 3];   // depth d+1

#pragma unroll
    for (int q = 0; q < OUTS; ++q) {
      const int i  = t + q * NTHREADS;
      const int hl = i / Ww;
      const int wl = i % Ww;
      const int c  = (hl + 1) * HWP + (wl + 1);

      // previous depth plane: full 3x3
      float m;
      m = fmaxf(fmaxf(pp[c - HWP - 1], pp[c - HWP]), pp[c - HWP + 1]);
      m = fmaxf(m, fmaxf(fmaxf(pp[c - 1], pp[c]), pp[c + 1]));
      m = fmaxf(m, fmaxf(fmaxf(pp[c + HWP - 1], pp[c + HWP]), pp[c + HWP + 1]));
      // next depth plane: full 3x3
      m = fmaxf(m, fmaxf(fmaxf(pn[c - HWP - 1], pn[c - HWP]), pn[c - HWP + 1]));
      m = fmaxf(m, fmaxf(fmaxf(pn[c - 1], pn[c]), pn[c + 1]));
      m = fmaxf(m, fmaxf(fmaxf(pn[c + HWP - 1], pn[c + HWP]), pn[c + HWP + 1]));
      // current plane: 8 neighbors, center excluded
      m = fmaxf(m, fmaxf(fmaxf(pc[c - HWP - 1], pc[c - HWP]), pc[c - HWP + 1]));
      m = fmaxf(m, fmaxf(pc[c - 1], pc[c + 1]));
      m = fmaxf(m, fmaxf(fmaxf(pc[c + HWP - 1], pc[c + HWP]), pc[c + HWP + 1]));
      // reference initializes the running max with zeros
      m = fmaxf(m, 0.0f);

      const float v = pc[c];
      const float r = (v > m) ? v : 0.0f;
      __builtin_nontemporal_store(r, &os[(size_t)d * (Hh * Ww) + (size_t)(h0 + hl) * Ww + wl]);
    }
    __syncthreads();   // all reads done before next iteration's DMA overwrites
  }
}

extern "C" void kernel_launch(void* const* d_in, const int* in_sizes, int n_in,
                              void* d_out, int out_size, void* d_ws, size_t ws_size,
                              hipStream_t stream) {
  (void)n_in; (void)out_size; (void)d_ws; (void)ws_size;
  const float* x = (const float*)d_in[0];
  float* out = (float*)d_out;

  const int slices = in_sizes[0] / (Dd * Hh * Ww);   // B*CH = 64
  const dim3 grid(slices * NTD * NTH);               // 1280 blocks
  const dim3 block(NTHREADS);                        // 8 wave32s
  nms3d_kernel<<<grid, block, 0, stream>>>(x, out);
}